// RoiPooling_26207890440586
// MI455X (gfx1250) — compile-verified
//
#include <hip/hip_runtime.h>
#include <cstdint>
#include <cstddef>

// Problem constants (match the reference)
#define RR        32
#define NN        20000
#define CC        128
#define OUTSZ     8
#define CELLS     (OUTSZ * OUTSZ)       // 64
#define POOL_ELEMS (CELLS * CC)         // 8192 floats = 32KB
#define TILE      2048
#define THREADS   1024

__global__ __launch_bounds__(THREADS)
void roi_pool_max_kernel(const float* __restrict__ feats,
                         const int*   __restrict__ coords,
                         const int*   __restrict__ batch_ids,
                         const int*   __restrict__ roi_boxes,
                         float*       __restrict__ out)
{
    __shared__ __align__(16) float pool[POOL_ELEMS];  // 32KB accumulator
    __shared__ int list_n[TILE];                      // compacted point idx
    __shared__ int list_cell[TILE];                   // compacted cell idx
    __shared__ int cnt;

    const int r   = blockIdx.x;
    const int tid = threadIdx.x;

    // ROI parameters (uniform -> scalar regs; division denominators uniform
    // so the backend hoists the exact-int-div rcp sequence to SALU)
    const int rb  = roi_boxes[r * 5 + 0];
    const int lo0 = roi_boxes[r * 5 + 1];
    const int lo1 = roi_boxes[r * 5 + 2];
    const int hi0 = roi_boxes[r * 5 + 3];
    const int hi1 = roi_boxes[r * 5 + 4];
    const int sz0 = max(hi0 - lo0, 1);
    const int sz1 = max(hi1 - lo1, 1);

    // init accumulator to -inf
    for (int i = tid; i < POOL_ELEMS; i += THREADS)
        pool[i] = -__builtin_inff();
    if (tid == 0) cnt = 0;
    __syncthreads();

    const int wave   = tid >> 5;          // 0..31
    const int lane   = tid & 31;
    const int nwaves = THREADS / 32;

    // membership test + append; prefetch the feature row on hit
    auto test_point = [&](int n, int c0, int c1, int bid) {
        const bool inside = (bid == rb) &&
                            (c0 >= lo0) && (c0 < hi0) &&
                            (c1 >= lo1) && (c1 < hi1);
        if (inside) {
            // rel in [0, size) -> trunc == floor, matches jnp //
            const int b0   = min(((c0 - lo0) * OUTSZ) / sz0, OUTSZ - 1);
            const int b1   = min(((c1 - lo1) * OUTSZ) / sz1, OUTSZ - 1);
            const int cell = b0 * OUTSZ + b1;
            const int slot = atomicAdd(&cnt, 1);     // ds_add_rtn_u32
            list_n[slot]    = n;
            list_cell[slot] = cell;
            const float* row = feats + (size_t)n * CC;
            __builtin_prefetch(row, 0, 0);           // global_prefetch_b8
            __builtin_prefetch(row + 64, 0, 0);      // +256B half
        }
    };

    for (int base = 0; base < NN; base += TILE) {
        const int tcount = min(TILE, NN - base);

        // ---- Stage 1: 2 points per thread, one wide load pair per thread ----
        for (int i = tid * 2; i < tcount; i += THREADS * 2) {
            const int n   = base + i;                         // even
            const int4 c4 = *(const int4*)(coords + (size_t)n * 2); // 16B
            const int2 b2 = *(const int2*)(batch_ids + n);          // 8B
            test_point(n, c4.x, c4.y, b2.x);
            if (i + 1 < tcount) test_point(n + 1, c4.z, c4.w, b2.y);
        }
        __syncthreads();
        const int m = cnt;

        // ---- Stage 2: one wave per listed point; coalesced 512B row gather,
        //      per-channel LDS atomic max (ds_max_num_f32, non-returning) ----
        for (int s = wave; s < m; s += nwaves) {
            const int n    = list_n[s];
            const int cell = list_cell[s];
            const float4 v = *(const float4*)(feats + (size_t)n * CC + lane * 4);
            const unsigned lds =
                (unsigned)(uintptr_t)(&pool[cell * CC + lane * 4]);
            asm volatile("ds_max_num_f32 %0, %1\n\t"
                         "ds_max_num_f32 %0, %2 offset:4\n\t"
                         "ds_max_num_f32 %0, %3 offset:8\n\t"
                         "ds_max_num_f32 %0, %4 offset:12"
                         :: "v"(lds), "v"(v.x), "v"(v.y), "v"(v.z), "v"(v.w)
                         : "memory");
        }
        __syncthreads();
        if (tid == 0) cnt = 0;
        __syncthreads();
    }

    // inline-asm DS atomics are invisible to the compiler's DScnt tracking:
    // drain them before cross-wave reads of pool[]
    asm volatile("s_wait_dscnt 0x0" ::: "memory");
    __syncthreads();

    // ---- Finalize in LDS: empty cells (-inf) -> 0 ----
    for (int i = tid; i < POOL_ELEMS; i += THREADS) {
        const float v = pool[i];
        pool[i] = (v == -__builtin_inff()) ? 0.0f : v;
    }
    asm volatile("s_wait_dscnt 0x0" ::: "memory");
    __syncthreads();

    // ---- Async LDS -> global store of the 32KB result tile ----
    // Each lane DMAs 16B straight out of LDS (ASYNCcnt-tracked, no VGPR trip).
    float* dst = out + (size_t)r * POOL_ELEMS;
    for (int i = tid * 4; i < POOL_ELEMS; i += THREADS * 4) {
        const unsigned lds = (unsigned)(uintptr_t)(&pool[i]);
        const uint64_t ga  = (uint64_t)(uintptr_t)(dst + i);
        asm volatile("global_store_async_from_lds_b128 %0, %1, off"
                     :: "v"(ga), "v"(lds) : "memory");
    }
    asm volatile("s_wait_asynccnt 0x0" ::: "memory");
}

extern "C" void kernel_launch(void* const* d_in, const int* in_sizes, int n_in,
                              void* d_out, int out_size, void* d_ws, size_t ws_size,
                              hipStream_t stream)
{
    const float* feats     = (const float*)d_in[0];  // (N, C) f32
    const int*   coords    = (const int*)  d_in[1];  // (N, 2) i32
    const int*   batch_ids = (const int*)  d_in[2];  // (N,)   i32
    const int*   roi_boxes = (const int*)  d_in[3];  // (R, 5) i32
    float*       out       = (float*)d_out;          // (R,8,8,C) f32

    roi_pool_max_kernel<<<dim3(RR), dim3(THREADS), 0, stream>>>(
        feats, coords, batch_ids, roi_boxes, out);
}